// LinearAutoDecoder_83468394430874
// MI455X (gfx1250) — compile-verified
//
#include <hip/hip_runtime.h>

// LinearAutoDecoder: out[n, 0:3] = dot(X[n, :127], Wcat[3*cid[n] + ch, :127])
// Wcat = [W_pos (768x63) | W_feat (768x64)].
// Memory-bound: 133 MB X stream @ 23.3 TB/s ~ 5.9 us floor. WMMA f32 16x16x4
// computes 16 per-row dot products per tile as the diagonal of a 16x16 tile
// product (16x MAC overcount = 3.2 GFLOP total, free vs. the HBM floor, exact
// fp32 accumulation). Weights are pre-merged into a padded 768x128 array in
// d_ws so the B-fragment gather is a single unconditional dword load per
// element (no exec-mask branches in the hot loop).

#define KTOT   127   // 63 pos + 64 latent
#define INFEAT 63
#define LATF   64
#define NROWS  768   // 3 * 256 clusters
#define WPAD   128   // merged weight row stride (col 127 zeroed)
#define TILE   16
#define XPAD   132   // LDS row stride: 132 mod 64 == 4 -> conflict-free column reads

typedef __attribute__((ext_vector_type(2))) float v2f;
typedef __attribute__((ext_vector_type(8))) float v8f;

// ---------------------------------------------------------------------------
// Prep: Wc[row][k] = k<63 ? W_pos[row][k] : (k<127 ? W_feat[row][k-63] : 0)
// 768*128 = 98304 elements; ~0.8 MB traffic, negligible.
// ---------------------------------------------------------------------------
__global__ __launch_bounds__(256)
void merge_weights_kernel(const float* __restrict__ Wp,
                          const float* __restrict__ Wf,
                          float*       __restrict__ Wc)
{
    int i = blockIdx.x * 256 + threadIdx.x;     // 0 .. 98303
    int row = i >> 7;
    int k   = i & 127;
    float v;
    if (k < INFEAT)      v = Wp[row * INFEAT + k];
    else if (k < KTOT)   v = Wf[row * LATF + (k - INFEAT)];
    else                 v = 0.0f;
    Wc[i] = v;
}

// ---------------------------------------------------------------------------
// Fast path: merged padded weights in Wc.
// ---------------------------------------------------------------------------
__global__ __launch_bounds__(32)
void LinearAutoDecoder_83468394430874_kernel(const float* __restrict__ X,
                                             const int*   __restrict__ cid,
                                             const float* __restrict__ Wc,
                                             float*       __restrict__ out)
{
    __shared__ float xs[TILE * XPAD];   // X tile, 16 rows x 128 (K=127 zero-padded)
    __shared__ float dg[3 * 32 * 8];    // accumulator dump for diagonal extraction

    const int lane = threadIdx.x;           // wave32: 0..31
    const int r0   = blockIdx.x * TILE;     // 16 rows per wave/tile

    // ---- Stage X tile into LDS: 2032 contiguous floats, fully coalesced ----
    const float* xg = X + (size_t)r0 * KTOT;
    for (int e = lane; e < TILE * KTOT; e += 32) {
        unsigned row = (unsigned)e / KTOT;
        unsigned col = (unsigned)e - row * KTOT;
        xs[row * XPAD + col] = xg[e];
    }
    if (lane < TILE) xs[lane * XPAD + KTOT] = 0.0f;   // zero-pad K=127
    __syncthreads();

    // ---- Per-lane gathered weight row bases (B column N = lane & 15) ----
    const int n  = lane & 15;
    const int hl = lane >> 4;                // half-wave selector
    const int wr = 3 * cid[r0 + n];          // cluster's first weight row
    // Per-lane/channel base pointers; hot-loop address = w?[4*s] / w?[4*s+2].
    const float* w0 = Wc + (size_t)(wr + 0) * WPAD + hl;
    const float* w1 = Wc + (size_t)(wr + 1) * WPAD + hl;
    const float* w2 = Wc + (size_t)(wr + 2) * WPAD + hl;
    const float* xa = xs + n * XPAD + 2 * hl;

    v8f acc0 = {}; v8f acc1 = {}; v8f acc2 = {};

    // ---- K loop: 32 steps of V_WMMA_F32_16X16X4_F32 per channel ----
    #pragma unroll 4
    for (int s = 0; s < 32; ++s) {
        // A 16x4 fp32 fragment: lanes 0-15 K=4s..4s+1 (M=lane),
        // lanes 16-31 K=4s+2..4s+3 (M=lane-16). Conflict-free LDS b64 reads.
        v2f a;
        a.x = xa[4 * s];
        a.y = xa[4 * s + 1];

        // B 4x16 fp32 fragment: lane holds column N at K = 4s+hl and 4s+2+hl.
        // Padded Wc row makes k=127 a plain load of the stored zero.
        {
            v2f b; b.x = w0[4 * s]; b.y = w0[4 * s + 2];
            acc0 = __builtin_amdgcn_wmma_f32_16x16x4_f32(false, a, false, b,
                                                         (short)0, acc0, false, false);
        }
        {
            v2f b; b.x = w1[4 * s]; b.y = w1[4 * s + 2];
            acc1 = __builtin_amdgcn_wmma_f32_16x16x4_f32(false, a, false, b,
                                                         (short)0, acc1, false, false);
        }
        {
            v2f b; b.x = w2[4 * s]; b.y = w2[4 * s + 2];
            acc2 = __builtin_amdgcn_wmma_f32_16x16x4_f32(false, a, false, b,
                                                         (short)0, acc2, false, false);
        }
    }

    // ---- Diagonal extraction: D[r][r] is the dot product for row r0+r ----
    // C/D layout: VGPR v -> lanes 0-15 (M=v, N=lane); lanes 16-31 (M=v+8, N=lane-16).
    #pragma unroll
    for (int j = 0; j < 8; ++j) {
        dg[0 * 256 + lane * 8 + j] = acc0[j];
        dg[1 * 256 + lane * 8 + j] = acc1[j];
        dg[2 * 256 + lane * 8 + j] = acc2[j];
    }
    __syncthreads();

    if (lane < TILE) {
        const int v   = lane & 7;
        const int src = (lane < 8) ? lane : (lane + 16);
        float* o = out + (size_t)(r0 + lane) * 3;
        o[0] = dg[0 * 256 + src * 8 + v];
        o[1] = dg[1 * 256 + src * 8 + v];
        o[2] = dg[2 * 256 + src * 8 + v];
    }
}

// ---------------------------------------------------------------------------
// Fallback (ws too small): branchless split-weight gather. Address select via
// cndmask + unconditional load + value select; no exec-mask branches.
// ---------------------------------------------------------------------------
__device__ __forceinline__ float wgather(const float* bp, const float* bf, int k)
{
    int kc = k < 126 ? k : 126;                    // clamp pad index in-bounds
    const float* p = (kc < INFEAT) ? (bp + kc) : (bf + kc);
    float v = *p;                                  // unconditional load
    return (k < KTOT) ? v : 0.0f;                  // pad k=127 -> 0
}

__global__ __launch_bounds__(32)
void LinearAutoDecoder_fallback_kernel(const float* __restrict__ X,
                                       const int*   __restrict__ cid,
                                       const float* __restrict__ Wp,
                                       const float* __restrict__ Wf,
                                       float*       __restrict__ out)
{
    __shared__ float xs[TILE * XPAD];
    __shared__ float dg[3 * 32 * 8];

    const int lane = threadIdx.x;
    const int r0   = blockIdx.x * TILE;

    const float* xg = X + (size_t)r0 * KTOT;
    for (int e = lane; e < TILE * KTOT; e += 32) {
        unsigned row = (unsigned)e / KTOT;
        unsigned col = (unsigned)e - row * KTOT;
        xs[row * XPAD + col] = xg[e];
    }
    if (lane < TILE) xs[lane * XPAD + KTOT] = 0.0f;
    __syncthreads();

    const int n  = lane & 15;
    const int hl = lane >> 4;
    const int wr = 3 * cid[r0 + n];
    const float* p0 = Wp + (size_t)(wr + 0) * INFEAT;
    const float* p1 = Wp + (size_t)(wr + 1) * INFEAT;
    const float* p2 = Wp + (size_t)(wr + 2) * INFEAT;
    const float* f0 = Wf + (size_t)(wr + 0) * LATF - INFEAT;  // so f0[k] = Wf[row][k-63]
    const float* f1 = Wf + (size_t)(wr + 1) * LATF - INFEAT;
    const float* f2 = Wf + (size_t)(wr + 2) * LATF - INFEAT;
    const float* xa = xs + n * XPAD + 2 * hl;

    v8f acc0 = {}; v8f acc1 = {}; v8f acc2 = {};

    #pragma unroll 4
    for (int s = 0; s < 32; ++s) {
        v2f a;
        a.x = xa[4 * s];
        a.y = xa[4 * s + 1];
        const int k0 = 4 * s + hl;
        const int k2 = k0 + 2;
        {
            v2f b; b.x = wgather(p0, f0, k0); b.y = wgather(p0, f0, k2);
            acc0 = __builtin_amdgcn_wmma_f32_16x16x4_f32(false, a, false, b,
                                                         (short)0, acc0, false, false);
        }
        {
            v2f b; b.x = wgather(p1, f1, k0); b.y = wgather(p1, f1, k2);
            acc1 = __builtin_amdgcn_wmma_f32_16x16x4_f32(false, a, false, b,
                                                         (short)0, acc1, false, false);
        }
        {
            v2f b; b.x = wgather(p2, f2, k0); b.y = wgather(p2, f2, k2);
            acc2 = __builtin_amdgcn_wmma_f32_16x16x4_f32(false, a, false, b,
                                                         (short)0, acc2, false, false);
        }
    }

    #pragma unroll
    for (int j = 0; j < 8; ++j) {
        dg[0 * 256 + lane * 8 + j] = acc0[j];
        dg[1 * 256 + lane * 8 + j] = acc1[j];
        dg[2 * 256 + lane * 8 + j] = acc2[j];
    }
    __syncthreads();

    if (lane < TILE) {
        const int v   = lane & 7;
        const int src = (lane < 8) ? lane : (lane + 16);
        float* o = out + (size_t)(r0 + lane) * 3;
        o[0] = dg[0 * 256 + src * 8 + v];
        o[1] = dg[1 * 256 + src * 8 + v];
        o[2] = dg[2 * 256 + src * 8 + v];
    }
}

extern "C" void kernel_launch(void* const* d_in, const int* in_sizes, int n_in,
                              void* d_out, int out_size, void* d_ws, size_t ws_size,
                              hipStream_t stream) {
    const float* X   = (const float*)d_in[0];   // (N, 127) fp32
    const int*   cid = (const int*)d_in[1];     // (N,) int32
    const float* Wp  = (const float*)d_in[2];   // (768, 63) fp32
    const float* Wf  = (const float*)d_in[3];   // (768, 64) fp32
    float*       out = (float*)d_out;           // (N, 3) fp32

    const int npts  = in_sizes[1];              // N = 262144
    const int tiles = npts / TILE;              // 16384 (N divisible by 16)

    const size_t wc_bytes = (size_t)NROWS * WPAD * sizeof(float);   // 393 KB
    if (ws_size >= wc_bytes) {
        float* Wc = (float*)d_ws;
        merge_weights_kernel<<<(NROWS * WPAD) / 256, 256, 0, stream>>>(Wp, Wf, Wc);
        LinearAutoDecoder_83468394430874_kernel<<<tiles, 32, 0, stream>>>(X, cid, Wc, out);
    } else {
        LinearAutoDecoder_fallback_kernel<<<tiles, 32, 0, stream>>>(X, cid, Wp, Wf, out);
    }
}